// MaskGenerate_23433341567066
// MI455X (gfx1250) — compile-verified
//
#include <hip/hip_runtime.h>

typedef __attribute__((ext_vector_type(2))) float v2f;
typedef __attribute__((ext_vector_type(8))) float v8f;

#define SEQ    197
#define MAP    (SEQ * SEQ)      // 38809 elements per attention map
#define LAYERS 12
#define HEADS  12
#define NRED   (LAYERS * HEADS) // 144 maps averaged
#define G      14
#define GP     (G * G)          // 196 patches (14x14 grid)
#define HW     784              // 28*28 output pixels
#define THRESH_PT 160           // on 28-grid areas; == 4 * area14

__global__ __launch_bounds__(256)
void maskgen_23433341567066_kernel(const float* __restrict__ att,
                                   float* __restrict__ out, int B)
{
    const int b   = blockIdx.x;
    const int tid = threadIdx.x;

    __shared__ float aP[208];      // per-patch mean (13 tiles x 16, padded)
    __shared__ float red[256];
    __shared__ int   msk[GP];
    __shared__ int   lab[GP];
    __shared__ int   cnt[GP];
    __shared__ int   changed;
    __shared__ int   bestLab;
    __shared__ int   keepFlag;
    __shared__ float thrS;

    // element strides for att[l][b][h][q][k]
    const size_t HSTR = (size_t)MAP;
    const size_t BSTR = (size_t)HEADS * MAP;
    const size_t LSTR = (size_t)B * BSTR;
    const float* base = att + (size_t)b * BSTR;

    // ---- Stage A (wave 0): 144-way mean of the CLS row via WMMA f32 16x16x4 ----
    // D = ones(16x4) * data(4x16) + C : every row of D is the column sum over K.
    // K = 12*l + (4*c2 + {hi, hi+2});  head offsets become constant 24-bit
    // immediate offsets on global_load_b32 (no div/mod address math).
    if (tid < 32) {
        const int lane = tid;
        const int n    = lane & 15;   // N column (patch within tile)
        const int hi   = lane >> 4;   // which K within the VGPR pair
        const v2f ones = {1.0f, 1.0f};
        for (int t = 0; t < 13; ++t) {
            const int p = t * 16 + n;           // padded patch index 0..207
            v8f acc = {0.f, 0.f, 0.f, 0.f, 0.f, 0.f, 0.f, 0.f};
            // per-lane pointer; hi folded in so remaining offsets are constants
            const float* pl = base + (size_t)hi * HSTR + (1 + p);
            for (int l = 0; l < LAYERS; ++l) {
#pragma unroll
                for (int c2 = 0; c2 < 3; ++c2) {
                    v2f bmat;
                    bmat.x = pl[(size_t)(4 * c2)     * HSTR];  // h = 4*c2 + hi
                    bmat.y = pl[(size_t)(4 * c2 + 2) * HSTR];  // h = 4*c2 + hi + 2
                    acc = __builtin_amdgcn_wmma_f32_16x16x4_f32(
                            false, ones, false, bmat, (short)0, acc, false, false);
                }
                pl += LSTR;
            }
            // D(0, n) lives in VGPR0 of lanes 0..15
            if (lane < 16 && p < GP) aP[p] = acc[0] * (1.0f / (float)NRED);
        }
    }
    __syncthreads();

    // ---- Stage B: per-image threshold = mean over patches (upsample-invariant) ----
    red[tid] = (tid < GP) ? aP[tid] : 0.0f;
    __syncthreads();
    for (int s = 128; s > 0; s >>= 1) {
        if (tid < s) red[tid] += red[tid + s];
        __syncthreads();
    }
    if (tid == 0) thrS = red[0] / (float)GP;
    __syncthreads();

    // ---- Stage C: mask + label seed (14-grid; order-consistent with 28-grid) ----
    if (tid < GP) {
        int m = (aP[tid] > thrS) ? 1 : 0;
        msk[tid] = m;
        lab[tid] = m ? tid : GP;   // background sentinel
    }
    __syncthreads();

    // ---- Stage D: 4-connected min-label propagation (Jacobi, converged exit) ----
    for (int it = 0; it < GP; ++it) {
        if (tid == 0) changed = 0;
        __syncthreads();
        int v = GP, vn = GP;
        if (tid < GP) {
            v = lab[tid];
            vn = v;
            if (msk[tid]) {
                const int y = tid / G, x = tid % G;
                int m = v;
                if (x > 0)     m = min(m, lab[tid - 1]);
                if (x < G - 1) m = min(m, lab[tid + 1]);
                if (y > 0)     m = min(m, lab[tid - G]);
                if (y < G - 1) m = min(m, lab[tid + G]);
                vn = m;
            }
        }
        __syncthreads();
        if (tid < GP) {
            lab[tid] = vn;
            if (vn != v) changed = 1;
        }
        __syncthreads();
        const int c = changed;
        __syncthreads();            // protect 'changed' read vs next-iter reset
        if (!c) break;
    }

    // ---- Stage E: component areas + argmax (first max == smallest label) ----
    if (tid < GP) cnt[tid] = 0;
    __syncthreads();
    if (tid < GP && msk[tid]) atomicAdd(&cnt[lab[tid]], 1);
    __syncthreads();
    if (tid == 0) {
        int bc = -1, bl = 0;
        for (int i = 0; i < GP; ++i)
            if (cnt[i] > bc) { bc = cnt[i]; bl = i; }
        bestLab  = bl;
        keepFlag = (4 * bc) >= THRESH_PT;
    }
    __syncthreads();

    // ---- Stage F: final mask (largest component if big enough, else raw mask) ----
    int fin = 0;
    if (tid < GP) fin = keepFlag ? (lab[tid] == bestLab) : msk[tid];
    __syncthreads();
    if (tid < GP) msk[tid] = fin;
    __syncthreads();

    // ---- Stage G: stable order of 28x28 flat indices (true first), +1 ----
    if (tid == 0) {
        float* o = out + (size_t)b * HW;
        int T = 0;
        for (int i = 0; i < HW; ++i) {
            const int y = i / 28, x = i % 28;
            T += msk[(y >> 1) * G + (x >> 1)];
        }
        int tc = 0, fc = 0;
        for (int i = 0; i < HW; ++i) {
            const int y = i / 28, x = i % 28;
            if (msk[(y >> 1) * G + (x >> 1)]) o[tc++]      = (float)(i + 1);
            else                              o[T + fc++]  = (float)(i + 1);
        }
    }
}

extern "C" void kernel_launch(void* const* d_in, const int* in_sizes, int n_in,
                              void* d_out, int out_size, void* d_ws, size_t ws_size,
                              hipStream_t stream) {
    const float* att = (const float*)d_in[0];
    const long long total = (long long)in_sizes[0];
    const int B = (int)(total / ((long long)LAYERS * HEADS * MAP));  // 32
    float* out = (float*)d_out;
    maskgen_23433341567066_kernel<<<dim3(B), dim3(256), 0, stream>>>(att, out, B);
}